// ListMLELoss_16381005267460
// MI455X (gfx1250) — compile-verified
//
#include <hip/hip_runtime.h>
#include <hip/hip_bf16.h>

#define ROW_LEN 2048
#define NT 256
#define EPT (ROW_LEN / NT)  // 8 elements per thread (aligned 8-blocks)

// ---- CDNA5 async global->LDS path (ASYNCcnt), with portable fallback ----
#if defined(__gfx1250__) && __has_builtin(__builtin_amdgcn_global_load_async_to_lds_b128)
#define HAVE_ASYNC_LDS 1
typedef int v4i __attribute__((ext_vector_type(4)));
typedef __attribute__((address_space(1))) v4i GlobalV4i;  // "__device__" in diagnostics
typedef __attribute__((address_space(3))) v4i LdsV4i;     // "__shared__" in diagnostics
#else
#define HAVE_ASYNC_LDS 0
#endif

#if defined(__gfx1250__) && __has_builtin(__builtin_amdgcn_s_wait_asynccnt)
#define WAIT_ASYNCCNT0() __builtin_amdgcn_s_wait_asynccnt(0)
#else
#define WAIT_ASYNCCNT0()
#endif

// Copy 32 bytes (8 floats) from global to LDS for this lane.
__device__ __forceinline__ void stage_32B(const float* __restrict__ gsrc,
                                          float* __restrict__ ldst) {
#if HAVE_ASYNC_LDS
  GlobalV4i* g = (GlobalV4i*)gsrc;
  LdsV4i* l = (LdsV4i*)ldst;
  __builtin_amdgcn_global_load_async_to_lds_b128(g, l, 0, 0);
  __builtin_amdgcn_global_load_async_to_lds_b128(g + 1, l + 1, 0, 0);
#else
  const float4* g = (const float4*)gsrc;
  float4* l = (float4*)ldst;
  l[0] = g[0];
  l[1] = g[1];
#endif
}

// Order-preserving float -> uint key transform (handles sign correctly).
__device__ __forceinline__ unsigned int fkey(unsigned int u) {
  return u ^ ((u >> 31) ? 0xFFFFFFFFu : 0x80000000u);
}

// Compare-exchange: descFirst=true keeps the LARGER key in the first slot.
__device__ __forceinline__ void cswap(unsigned int& ka, float& va,
                                      unsigned int& kb, float& vb,
                                      bool descFirst) {
  bool sw = descFirst ? (ka < kb) : (ka > kb);
  if (sw) {
    unsigned int kt = ka; ka = kb; kb = kt;
    float vt = va; va = vb; vb = vt;
  }
}

// Merge a bitonic 8-sequence held in registers (sub-stages j=4,2,1),
// single direction d for all pairs.
__device__ __forceinline__ void local_merge8(unsigned int k8[EPT], float v8[EPT], bool d) {
#pragma unroll
  for (int c = 0; c < 4; ++c) cswap(k8[c], v8[c], k8[c + 4], v8[c + 4], d);
#pragma unroll
  for (int h = 0; h < 8; h += 4) {
    cswap(k8[h + 0], v8[h + 0], k8[h + 2], v8[h + 2], d);
    cswap(k8[h + 1], v8[h + 1], k8[h + 3], v8[h + 3], d);
  }
#pragma unroll
  for (int h = 0; h < 8; h += 2) cswap(k8[h], v8[h], k8[h + 1], v8[h + 1], d);
}

// Full in-register bitonic network for stages k=2,4,8 on an aligned 8-block.
// Directions follow dir(i) = ((i & k) == 0) with i = t*8 + c; base bits < 8 vanish.
__device__ __forceinline__ void local_sort8(unsigned int k8[EPT], float v8[EPT], bool d8) {
  // k=2
  cswap(k8[0], v8[0], k8[1], v8[1], true);
  cswap(k8[2], v8[2], k8[3], v8[3], false);
  cswap(k8[4], v8[4], k8[5], v8[5], true);
  cswap(k8[6], v8[6], k8[7], v8[7], false);
  // k=4, j=2
  cswap(k8[0], v8[0], k8[2], v8[2], true);
  cswap(k8[1], v8[1], k8[3], v8[3], true);
  cswap(k8[4], v8[4], k8[6], v8[6], false);
  cswap(k8[5], v8[5], k8[7], v8[7], false);
  // k=4, j=1
  cswap(k8[0], v8[0], k8[1], v8[1], true);
  cswap(k8[2], v8[2], k8[3], v8[3], true);
  cswap(k8[4], v8[4], k8[5], v8[5], false);
  cswap(k8[6], v8[6], k8[7], v8[7], false);
  // k=8 (j=4,2,1), direction = ((t & 1) == 0)
  local_merge8(k8, v8, d8);
}

__global__ __launch_bounds__(NT) void listmle_row_kernel(
    const float* __restrict__ scores, const float* __restrict__ labels,
    float* __restrict__ row_loss) {
  __shared__ __align__(16) unsigned int kkey[ROW_LEN];  // sort keys (label bits)
  __shared__ __align__(16) float sval[ROW_LEN];         // payload (scores)
  __shared__ float sc[NT];
  __shared__ float red[NT];

  const int row = blockIdx.x;
  const int t = threadIdx.x;
  const size_t base = (size_t)row * ROW_LEN + (size_t)t * EPT;

  // ---- Stage row into LDS (async global->LDS data movement, ASYNCcnt) ----
  stage_32B(labels + base, (float*)&kkey[t * EPT]);
  stage_32B(scores + base, &sval[t * EPT]);
  WAIT_ASYNCCNT0();
  __syncthreads();

  // ---- Phase A: in-register bitonic stages k=2,4,8 on this thread's 8-block ----
  unsigned int k8[EPT];
  float v8[EPT];
#pragma unroll
  for (int c = 0; c < EPT; ++c) {
    k8[c] = fkey(kkey[t * EPT + c]);
    v8[c] = sval[t * EPT + c];
  }
  local_sort8(k8, v8, (t & 1) == 0);
#pragma unroll
  for (int c = 0; c < EPT; ++c) {
    kkey[t * EPT + c] = k8[c];
    sval[t * EPT + c] = v8[c];
  }
  __syncthreads();

  // ---- Phase B: stages k=16..2048; LDS passes for j>=8, registers for j=4,2,1 ----
  for (unsigned int k = 2 * EPT; k <= ROW_LEN; k <<= 1) {
    for (unsigned int j = k >> 1; j >= EPT; j >>= 1) {
#pragma unroll
      for (int rep = 0; rep < (ROW_LEN / 2) / NT; ++rep) {
        unsigned int idx = (unsigned int)t + (unsigned int)rep * NT;
        unsigned int i = ((idx & ~(j - 1u)) << 1) | (idx & (j - 1u));
        unsigned int p = i | j;
        bool descBlock = ((i & k) == 0u);
        unsigned int ka = kkey[i], kb = kkey[p];
        bool doSwap = descBlock ? (ka < kb) : (ka > kb);
        if (doSwap) {
          kkey[i] = kb;
          kkey[p] = ka;
          float va = sval[i];
          sval[i] = sval[p];
          sval[p] = va;
        }
      }
      __syncthreads();
    }
    // Thread-local tail of this merge: j=4,2,1; direction constant per 8-block.
    const bool d = ((((unsigned int)t * EPT) & k) == 0u);
#pragma unroll
    for (int c = 0; c < EPT; ++c) {
      k8[c] = kkey[t * EPT + c];
      v8[c] = sval[t * EPT + c];
    }
    local_merge8(k8, v8, d);
    if (k < ROW_LEN) {  // after the final merge the sorted block stays in registers
#pragma unroll
      for (int c = 0; c < EPT; ++c) {
        kkey[t * EPT + c] = k8[c];
        sval[t * EPT + c] = v8[c];
      }
      __syncthreads();
    }
  }
  // v8[] now holds this thread's 8 scores in globally sorted (desc-by-label) order.

  // ---- Row max over scores (single stable logsumexp shift) ----
  float lm = -INFINITY;
#pragma unroll
  for (int c = 0; c < EPT; ++c) lm = fmaxf(lm, v8[c]);
  red[t] = lm;
  __syncthreads();
  for (int d = NT / 2; d > 0; d >>= 1) {
    if (t < d) red[t] = fmaxf(red[t], red[t + d]);
    __syncthreads();
  }
  const float m = red[0];
  __syncthreads();

  // ---- Local reverse suffix sums of exp(s - m) within this thread's chunk ----
  float loc[EPT];
  float acc = 0.0f;
#pragma unroll
  for (int c = EPT - 1; c >= 0; --c) {
    acc += __expf(v8[c] - m);
    loc[c] = acc;  // sum over [c .. EPT) of this chunk
  }

  // ---- Reverse inclusive scan of the 256 chunk totals (Hillis-Steele) ----
  sc[t] = acc;
  __syncthreads();
  float v = acc;
  for (int d = 1; d < NT; d <<= 1) {
    float other = (t + d < NT) ? sc[t + d] : 0.0f;
    __syncthreads();
    v += other;
    sc[t] = v;
    __syncthreads();
  }
  const float offRight = v - acc;  // exp-sum of all chunks strictly right of t

  // ---- Per-thread partials: sum log(suffix_sum) and sum of scores ----
  float sum_log = 0.0f;
  float sum_s = 0.0f;
#pragma unroll
  for (int c = 0; c < EPT; ++c) {
    sum_log += __logf(loc[c] + offRight);
    sum_s += v8[c];
  }

  // ---- Block reduce both partials (deterministic tree) ----
  red[t] = sum_log;
  sc[t] = sum_s;
  __syncthreads();
  for (int d = NT / 2; d > 0; d >>= 1) {
    if (t < d) {
      red[t] += red[t + d];
      sc[t] += sc[t + d];
    }
    __syncthreads();
  }

  if (t == 0) {
    // sum_i (m + log(suffsum_i)) - sum_i s_i
    row_loss[row] = red[0] + (float)ROW_LEN * m - sc[0];
  }
}

__global__ __launch_bounds__(NT) void reduce_mean_kernel(
    const float* __restrict__ row_loss, float* __restrict__ out, int rows) {
  __shared__ float red[NT];
  const int t = threadIdx.x;
  float s = 0.0f;
  for (int i = t; i < rows; i += NT) s += row_loss[i];
  red[t] = s;
  __syncthreads();
  for (int d = NT / 2; d > 0; d >>= 1) {
    if (t < d) red[t] += red[t + d];
    __syncthreads();
  }
  if (t == 0) out[0] = red[0] / (float)rows;
}

extern "C" void kernel_launch(void* const* d_in, const int* in_sizes, int n_in,
                              void* d_out, int out_size, void* d_ws, size_t ws_size,
                              hipStream_t stream) {
  const float* scores = (const float*)d_in[0];
  const float* labels = (const float*)d_in[1];
  float* ws = (float*)d_ws;   // rows floats of per-row loss
  float* out = (float*)d_out; // single float

  const int rows = in_sizes[0] / ROW_LEN;  // B = 8192 for the reference shapes

  listmle_row_kernel<<<rows, NT, 0, stream>>>(scores, labels, ws);
  reduce_mean_kernel<<<1, NT, 0, stream>>>(ws, out, rows);
}